// GraphGATBridge_73778948211151
// MI455X (gfx1250) — compile-verified
//
#include <hip/hip_runtime.h>
#include <math.h>

// ---------------------------------------------------------------------------
// Types for V_WMMA_F32_16X16X4_F32 (fp32 WMMA, CDNA5/gfx1250)
// A: 16x4 f32 (2 VGPR), B: 4x16 f32 (2 VGPR), C/D: 16x16 f32 (8 VGPR)
// ---------------------------------------------------------------------------
typedef float v2f __attribute__((ext_vector_type(2)));
typedef float v8f __attribute__((ext_vector_type(8)));

__device__ __forceinline__ v8f wmma4(v2f a, v2f b, v8f c) {
  // 8 args: (neg_a, A, neg_b, B, c_mod, C, reuse_a, reuse_b)
  return __builtin_amdgcn_wmma_f32_16x16x4_f32(false, a, false, b,
                                               (short)0, c, false, false);
}

// Problem constants
#define Bn 16
#define Cc 256
#define Hh 128
#define Ww 128
#define Rr 64
#define Nn 64          // 8x8 grid nodes
#define PLANE (Hh*Ww)  // 16384

// ---------------------------------------------------------------------------
// Kernel 1: 16x16 block-mean pool of x -> xsA[(b*64+n)*256 + c]
// One workgroup per (b,c) plane; float4 loads; conflict-free reduction.
// ---------------------------------------------------------------------------
__global__ __launch_bounds__(256) void pool_kernel(
    const float* __restrict__ x, float* __restrict__ xsA) {
  __shared__ float part[256][8];
  const int bc  = blockIdx.x;       // b*256 + c
  const int tid = threadIdx.x;
  const float4* plane = (const float4*)(x + (size_t)bc * PLANE);

  // thread t: fixed col-quad j = (t&31)>>2 ; iteration it covers node-row it>>1
  float acc[8];
#pragma unroll
  for (int i = 0; i < 8; ++i) acc[i] = 0.f;
#pragma unroll
  for (int it = 0; it < 16; ++it) {
    float4 v = plane[tid + it * 256];
    acc[it >> 1] += (v.x + v.y) + (v.z + v.w);
  }
#pragma unroll
  for (int nr = 0; nr < 8; ++nr) part[tid][nr] = acc[nr];
  __syncthreads();

  if (tid < 64) {
    const int nr = tid >> 3, j = tid & 7;   // node = nr*8 + j
    float s = 0.f;
#pragma unroll
    for (int g = 0; g < 8; ++g)
#pragma unroll
      for (int l = 0; l < 4; ++l) s += part[g * 32 + j * 4 + l][nr];
    const int b = bc >> 8, c = bc & 255;
    xsA[((size_t)(b * Nn + nr * 8 + j)) * Cc + c] = s * (1.f / 256.f);
  }
}

// ---------------------------------------------------------------------------
// Kernel 2: per-batch  reduce-GEMM -> GAT (4 heads) -> restore-GEMM
// All three GEMMs use v_wmma_f32_16x16x4_f32. One workgroup (8 waves)/batch.
// Output restg[(b*256+c)*64 + node]
// ---------------------------------------------------------------------------
__global__ __launch_bounds__(256) void fused_node_kernel(
    const float* __restrict__ xsA,       // [B*64, 256]
    const float* __restrict__ reduce_w,  // [64, 256]
    const float* __restrict__ lin_w,     // [256, 64]
    const float* __restrict__ att_src,   // [4, 64]
    const float* __restrict__ att_dst,   // [4, 64]
    const float* __restrict__ gat_bias,  // [64]
    const float* __restrict__ restore_w, // [256, 64]
    float* __restrict__ restg) {         // [B*256, 64]
  __shared__ float nodes_s[Nn][Rr];   // 16 KB
  __shared__ float xp_s[Nn][Rr];      // 16 KB (per-head projected feats)
  __shared__ float outacc[Nn][Rr];    // 16 KB (head-summed messages)
  __shared__ float as_s[Nn], ad_s[Nn];
  __shared__ float wsm[Nn][8];        // softmax weights per dst (<=5 edges)

  const int b     = blockIdx.x;
  const int tid   = threadIdx.x;
  const int wave  = tid >> 5;
  const int lane  = tid & 31;
  const int lrow  = lane & 15;   // M / N index inside fragment
  const int lhalf = lane >> 4;   // K-pair / M-half selector

  // ---- reduce GEMM: nodes[64,64] = xsA_b[64,256] @ reduce_w^T[256,64] ----
  {
    const int mtile = wave & 3;
    const int nt0   = (wave >> 2) * 2;
    const float* Ab = xsA + ((size_t)(b * Nn + mtile * 16 + lrow)) * Cc + lhalf * 2;
    const float* B0 = reduce_w + (size_t)(nt0 * 16 + lrow) * Cc + lhalf * 2;
    const float* B1 = reduce_w + (size_t)((nt0 + 1) * 16 + lrow) * Cc + lhalf * 2;
    v8f c0 = {}, c1 = {};
    for (int k0 = 0; k0 < Cc; k0 += 4) {
      v2f a  = *(const v2f*)(Ab + k0);
      v2f b0 = *(const v2f*)(B0 + k0);
      v2f b1 = *(const v2f*)(B1 + k0);
      c0 = wmma4(a, b0, c0);
      c1 = wmma4(a, b1, c1);
    }
#pragma unroll
    for (int v = 0; v < 8; ++v) {
      const int m = mtile * 16 + lhalf * 8 + v;
      nodes_s[m][nt0 * 16 + lrow]       = c0[v];
      nodes_s[m][(nt0 + 1) * 16 + lrow] = c1[v];
    }
  }
  for (int i = tid; i < Nn * Rr; i += 256) ((float*)outacc)[i] = 0.f;
  __syncthreads();

  // ---- GAT, head by head (softmax is independent per head) ----
  for (int h = 0; h < 4; ++h) {
    // xp = nodes @ lin_w_h^T  (M=64, N=64, K=64)
    {
      const int mtile = wave & 3;
      const int nt0   = (wave >> 2) * 2;
      const float* B0 = lin_w + (size_t)(h * 64 + nt0 * 16 + lrow) * Rr + lhalf * 2;
      const float* B1 = B0 + 16 * Rr;
      v8f c0 = {}, c1 = {};
      for (int k0 = 0; k0 < Rr; k0 += 4) {
        v2f a  = *(const v2f*)(&nodes_s[mtile * 16 + lrow][k0 + lhalf * 2]);
        v2f b0 = *(const v2f*)(B0 + k0);
        v2f b1 = *(const v2f*)(B1 + k0);
        c0 = wmma4(a, b0, c0);
        c1 = wmma4(a, b1, c1);
      }
#pragma unroll
      for (int v = 0; v < 8; ++v) {
        const int m = mtile * 16 + lhalf * 8 + v;
        xp_s[m][nt0 * 16 + lrow]       = c0[v];
        xp_s[m][(nt0 + 1) * 16 + lrow] = c1[v];
      }
    }
    __syncthreads();

    // attention logits a_s, a_d
    if (tid < Nn) {
      float s = 0.f, d = 0.f;
      for (int r = 0; r < Rr; ++r) {
        const float xv = xp_s[tid][r];
        s += xv * att_src[h * Rr + r];
        d += xv * att_dst[h * Rr + r];
      }
      as_s[tid] = s;
      ad_s[tid] = d;
    }
    __syncthreads();

    // segment softmax over incoming edges of each dst (4-neighborhood + self)
    if (tid < Nn) {
      const int i = tid >> 3, j = tid & 7;
      int srcs[5]; int ne = 0;
      if (i > 0) srcs[ne++] = tid - 8;
      if (i < 7) srcs[ne++] = tid + 8;
      if (j > 0) srcs[ne++] = tid - 1;
      if (j < 7) srcs[ne++] = tid + 1;
      srcs[ne++] = tid;  // self loop
      const float ad = ad_s[tid];
      float al[5], mx = -1e30f;
      for (int e = 0; e < ne; ++e) {
        float a = as_s[srcs[e]] + ad;
        a = (a > 0.f) ? a : 0.2f * a;   // leaky_relu 0.2
        al[e] = a;
        mx = fmaxf(mx, a);
      }
      float den = 0.f;
      for (int e = 0; e < ne; ++e) { al[e] = __expf(al[e] - mx); den += al[e]; }
      const float inv = 1.f / (den + 1e-16f);
      for (int e = 0; e < ne; ++e) wsm[tid][e] = al[e] * inv;
      for (int e = ne; e < 8; ++e) wsm[tid][e] = 0.f;
    }
    __syncthreads();

    // aggregate messages: outacc[n][r] += sum_e w_e * xp[src_e][r]
    {
      const int n  = tid >> 2;
      const int rb = (tid & 3) * 16;
      const int i = n >> 3, j = n & 7;
      int srcs[5]; int ne = 0;
      if (i > 0) srcs[ne++] = n - 8;
      if (i < 7) srcs[ne++] = n + 8;
      if (j > 0) srcs[ne++] = n - 1;
      if (j < 7) srcs[ne++] = n + 1;
      srcs[ne++] = n;
      for (int rr = 0; rr < 16; ++rr) {
        const int r = rb + rr;
        float acc = outacc[n][r];
        for (int e = 0; e < ne; ++e) acc += wsm[n][e] * xp_s[srcs[e]][r];
        outacc[n][r] = acc;
      }
    }
    __syncthreads();
  }

  // head mean + bias (in place)
  for (int idx = tid; idx < Nn * Rr; idx += 256) {
    const int n = idx >> 6, r = idx & 63;
    outacc[n][r] = outacc[n][r] * 0.25f + gat_bias[r];
  }
  __syncthreads();

  // ---- restore GEMM: rest[n][c] = sum_r outn[n][r]*restore_w[c][r] ----
  // D fragment: node runs along VGPR index -> contiguous b128 stores to restg
  {
    const int mtile = wave & 3;
    const int cbase = (wave >> 2) * 8;
    for (int ct = 0; ct < 8; ++ct) {
      const int ctile = cbase + ct;
      const float* Bp = restore_w + (size_t)(ctile * 16 + lrow) * Rr + lhalf * 2;
      v8f c0 = {};
      for (int k0 = 0; k0 < Rr; k0 += 4) {
        v2f a  = *(const v2f*)(&outacc[mtile * 16 + lrow][k0 + lhalf * 2]);
        v2f bb = *(const v2f*)(Bp + k0);
        c0 = wmma4(a, bb, c0);
      }
      const int c     = ctile * 16 + lrow;
      const int nbase = mtile * 16 + lhalf * 8;
      float* dst = restg + ((size_t)(b * Cc + c)) * Nn + nbase;
      *(float4*)(dst)     = make_float4(c0[0], c0[1], c0[2], c0[3]);
      *(float4*)(dst + 4) = make_float4(c0[4], c0[5], c0[6], c0[7]);
    }
  }
}

// ---------------------------------------------------------------------------
// Bilinear 8x8 -> 128x128 tap (align_corners=False / half-pixel, edge clamp)
// ---------------------------------------------------------------------------
__device__ __forceinline__ float bilin(const float* __restrict__ g, int hh, int ww) {
  const float sh = (hh + 0.5f) * (1.f / 16.f) - 0.5f;
  const float sw = (ww + 0.5f) * (1.f / 16.f) - 0.5f;
  const float fh0 = floorf(sh), fw0 = floorf(sw);
  const float fh = sh - fh0, fw = sw - fw0;
  const int h0 = max(0, min(7, (int)fh0));
  const int h1 = max(0, min(7, (int)fh0 + 1));
  const int w0 = max(0, min(7, (int)fw0));
  const int w1 = max(0, min(7, (int)fw0 + 1));
  const float v00 = g[h0 * 8 + w0], v01 = g[h0 * 8 + w1];
  const float v10 = g[h1 * 8 + w0], v11 = g[h1 * 8 + w1];
  const float t0 = v00 + fw * (v01 - v00);
  const float t1 = v10 + fw * (v11 - v10);
  return t0 + fh * (t1 - t0);
}

// ---------------------------------------------------------------------------
// Kernel 3: per-plane partial sums of pre = x + up  (deterministic reduction)
// partials[bc*2+0] = sum(pre), partials[bc*2+1] = sum(pre^2)
// ---------------------------------------------------------------------------
__global__ __launch_bounds__(256) void stats_kernel(
    const float* __restrict__ x, const float* __restrict__ restg,
    float* __restrict__ partials) {
  __shared__ float g[Nn];
  __shared__ float red[256];
  const int bc = blockIdx.x, tid = threadIdx.x;
  if (tid < Nn) g[tid] = restg[(size_t)bc * Nn + tid];
  __syncthreads();

  const float4* plane = (const float4*)(x + (size_t)bc * PLANE);
  float s1 = 0.f, s2 = 0.f;
  for (int it = 0; it < 16; ++it) {
    const int q = tid + it * 256;           // float4 index
    const float4 v = plane[q];
    const int row = q >> 5, col = (q & 31) * 4;
    const float p0 = v.x + bilin(g, row, col + 0);
    const float p1 = v.y + bilin(g, row, col + 1);
    const float p2 = v.z + bilin(g, row, col + 2);
    const float p3 = v.w + bilin(g, row, col + 3);
    s1 += (p0 + p1) + (p2 + p3);
    s2 += (p0 * p0 + p1 * p1) + (p2 * p2 + p3 * p3);
  }
  red[tid] = s1; __syncthreads();
  for (int off = 128; off > 0; off >>= 1) {
    if (tid < off) red[tid] += red[tid + off];
    __syncthreads();
  }
  if (tid == 0) partials[(size_t)bc * 2 + 0] = red[0];
  __syncthreads();
  red[tid] = s2; __syncthreads();
  for (int off = 128; off > 0; off >>= 1) {
    if (tid < off) red[tid] += red[tid + off];
    __syncthreads();
  }
  if (tid == 0) partials[(size_t)bc * 2 + 1] = red[0];
}

// Kernel 4: fold 16 batch-partials per channel -> sums[c*2 + {0,1}]
__global__ __launch_bounds__(256) void reduce_stats_kernel(
    const float* __restrict__ partials, float* __restrict__ sums) {
  const int c = threadIdx.x;  // 0..255
  float s1 = 0.f, s2 = 0.f;
  for (int b = 0; b < Bn; ++b) {
    s1 += partials[((size_t)(b * Cc + c)) * 2 + 0];
    s2 += partials[((size_t)(b * Cc + c)) * 2 + 1];
  }
  sums[c * 2 + 0] = s1;
  sums[c * 2 + 1] = s2;
}

// ---------------------------------------------------------------------------
// Kernel 5: out = relu(gamma*(x+up-mean)*rsqrt(var+eps)+beta)
// ---------------------------------------------------------------------------
__global__ __launch_bounds__(256) void finalize_kernel(
    const float* __restrict__ x, const float* __restrict__ restg,
    const float* __restrict__ sums, const float* __restrict__ gamma,
    const float* __restrict__ beta, float* __restrict__ out) {
  __shared__ float g[Nn];
  const int bc = blockIdx.x, tid = threadIdx.x;
  const int c = bc & 255;
  if (tid < Nn) g[tid] = restg[(size_t)bc * Nn + tid];
  __syncthreads();

  const float invN = 1.f / ((float)Bn * PLANE);
  const float mean = sums[c * 2 + 0] * invN;
  const float var  = sums[c * 2 + 1] * invN - mean * mean;
  const float istd = rsqrtf(var + 1e-5f);
  const float gm = gamma[c] * istd;
  const float bt = beta[c] - mean * gm;

  const float4* px = (const float4*)(x + (size_t)bc * PLANE);
  float4* po = (float4*)(out + (size_t)bc * PLANE);
  for (int it = 0; it < 16; ++it) {
    const int q = tid + it * 256;
    const float4 v = px[q];
    const int row = q >> 5, col = (q & 31) * 4;
    float4 o;
    o.x = fmaxf(0.f, gm * (v.x + bilin(g, row, col + 0)) + bt);
    o.y = fmaxf(0.f, gm * (v.y + bilin(g, row, col + 1)) + bt);
    o.z = fmaxf(0.f, gm * (v.z + bilin(g, row, col + 2)) + bt);
    o.w = fmaxf(0.f, gm * (v.w + bilin(g, row, col + 3)) + bt);
    po[q] = o;
  }
}

// ---------------------------------------------------------------------------
extern "C" void kernel_launch(void* const* d_in, const int* in_sizes, int n_in,
                              void* d_out, int out_size, void* d_ws, size_t ws_size,
                              hipStream_t stream) {
  const float* x         = (const float*)d_in[0];
  const float* reduce_w  = (const float*)d_in[1];
  const float* lin_w     = (const float*)d_in[2];
  const float* att_src   = (const float*)d_in[3];
  const float* att_dst   = (const float*)d_in[4];
  const float* gat_bias  = (const float*)d_in[5];
  const float* restore_w = (const float*)d_in[6];
  const float* bn_gamma  = (const float*)d_in[7];
  const float* bn_beta   = (const float*)d_in[8];

  float* ws       = (float*)d_ws;
  float* xsA      = ws;                       // [1024*256]  = 262144 floats
  float* restg    = ws + 262144;              // [4096*64]   = 262144 floats
  float* partials = ws + 524288;              // [4096*2]    = 8192 floats
  float* sums     = ws + 532480;              // [256*2]     = 512 floats

  pool_kernel<<<Bn * Cc, 256, 0, stream>>>(x, xsA);
  fused_node_kernel<<<Bn, 256, 0, stream>>>(xsA, reduce_w, lin_w, att_src,
                                            att_dst, gat_bias, restore_w, restg);
  stats_kernel<<<Bn * Cc, 256, 0, stream>>>(x, restg, partials);
  reduce_stats_kernel<<<1, 256, 0, stream>>>(partials, sums);
  finalize_kernel<<<Bn * Cc, 256, 0, stream>>>(x, restg, sums, bn_gamma,
                                               bn_beta, (float*)d_out);
}